// BinaryPathEncoder_15788299780444
// MI455X (gfx1250) — compile-verified
//
#include <hip/hip_runtime.h>

// ---------------------------------------------------------------------------
// BinaryPathEncoder on MI455X (gfx1250), f32 WMMA path.
//
//   result(pos) = W_{b0}^T W_{b1}^T ... over bits of pos (LSB first, leading 1
//   excluded), W_b = expm(P_b - P_b^T).
//
//   Tree DP: Tab[q-1] = result(q) for q in [1,512), built level by level:
//     Tab[0] = I;  Tab[q-1] = Wt[q&1] @ Tab[(q>>1)-1]   (Wt[b] = W_b^T)
//   Final:  pos >= 256: out = Tab[255 + (pos&255)] @ Tab[(pos>>8)-1]
//           pos <  256: out = Tab[pos-1] @ Tab[0]  (Tab[0] = I)
//
//   36 GFLOP of f32 WMMA total; 33.5 MB table stays L2-resident; HBM cost is
//   the mandatory 537 MB of non-temporal output stores (~23 us @ 23.3 TB/s).
// ---------------------------------------------------------------------------

typedef __attribute__((ext_vector_type(2))) float v2f;
typedef __attribute__((ext_vector_type(8))) float v8f;

#define MAT    128
#define MATSZ  (MAT * MAT)     // 16384 floats per matrix
#define LDA    132             // A-side LDS stride: 132%64==4 -> b64 A-frag
                               //   loads of 16 rows spread across all banks
#define LDB    144             // B-side LDS stride: 144%64==16 -> the two
                               //   16-lane halves hit disjoint bank sets
#define LDSE   144             // expm: single stride for all three matrices
#define LDSMAT_A (MAT * LDA)   // 16896 floats = 67584 B
#define LDSMAT_B (MAT * LDB)   // 18432 floats = 73728 B
#define LDSMAT_E (MAT * LDSE)

// ---- wave-level 32x128 @ 128x64 tile matmul (2x4 fragment grid) ------------
// A fragment (16x4 f32): lane m=lane&15, half=lane>>4:
//   a = { A[m][4ks+2h], A[m][4ks+2h+1] }        (contiguous b64 load)
// B fragment (4x16 f32): n=lane&15:
//   b = { B[4ks+2h][n], B[4ks+2h+1][n] }
// Software-pipelined: fragments for ks+1 are fetched before the WMMAs of ks.
__device__ __forceinline__ void wave_mm_2x4(
    const float* __restrict__ Ls, int ldL,
    const float* __restrict__ Rs, int ldR,
    int tmBase, int tnBase, int lane, v8f acc[8])
{
  const int n = lane & 15, half = lane >> 4;
  const float* Lr0 = Ls + (size_t)(16 * tmBase + n) * ldL + 2 * half;
  const float* Lr1 = Lr0 + (size_t)16 * ldL;
  const float* Rb  = Rs + (size_t)2 * half * ldR + 16 * tnBase + n;

  v2f a0, a1, b[4];
  a0.x = Lr0[0]; a0.y = Lr0[1];
  a1.x = Lr1[0]; a1.y = Lr1[1];
#pragma unroll
  for (int c = 0; c < 4; ++c) { b[c].x = Rb[16 * c]; b[c].y = Rb[16 * c + ldR]; }

#pragma unroll 4
  for (int ks = 0; ks < 32; ++ks) {
    // prefetch next iteration's fragments (ks==31 redundantly reloads 31)
    const int ks1 = (ks < 31) ? (ks + 1) : 31;
    const int off = 4 * ks1;
    v2f a0n, a1n, bn[4];
    a0n.x = Lr0[off]; a0n.y = Lr0[off + 1];
    a1n.x = Lr1[off]; a1n.y = Lr1[off + 1];
    const float* rb = Rb + (size_t)off * ldR;
#pragma unroll
    for (int c = 0; c < 4; ++c) { bn[c].x = rb[16 * c]; bn[c].y = rb[16 * c + ldR]; }

#pragma unroll
    for (int c = 0; c < 4; ++c) {
      acc[c]     = __builtin_amdgcn_wmma_f32_16x16x4_f32(
                     false, a0, false, b[c], (short)0, acc[c],     false, false);
      acc[4 + c] = __builtin_amdgcn_wmma_f32_16x16x4_f32(
                     false, a1, false, b[c], (short)0, acc[4 + c], false, false);
    }
    a0 = a0n; a1 = a1n;
#pragma unroll
    for (int c = 0; c < 4; ++c) b[c] = bn[c];
  }
}

// C/D layout: VGPR v, lane: row = v + 8*half, col = n  (within a 16x16 tile)
template <bool NT>
__device__ __forceinline__ void wave_store_2x4(
    float* D, int ldD, int tmBase, int tnBase, int lane, const v8f acc[8])
{
  const int n = lane & 15, half = lane >> 4;
#pragma unroll
  for (int r = 0; r < 2; ++r)
#pragma unroll
    for (int c = 0; c < 4; ++c) {
      float* dst = D + (size_t)(16 * (tmBase + r) + 8 * half) * ldD
                     + 16 * (tnBase + c) + n;
#pragma unroll
      for (int v = 0; v < 8; ++v) {
        if (NT) __builtin_nontemporal_store(acc[r * 4 + c][v],
                                            dst + (size_t)v * ldD);
        else    dst[(size_t)v * ldD] = acc[r * 4 + c][v];
      }
    }
}

// Stage a 128x128 row-major global matrix into LDS with row stride ldS.
__device__ __forceinline__ void stage_matrix(
    const float* __restrict__ G, float* S, int ldS, int tid)
{
  for (int idx = tid; idx < MAT * 32; idx += 256) {
    const int row = idx >> 5;
    const int c4  = (idx & 31) << 2;
    const float4 v = *(const float4*)(G + (size_t)row * MAT + c4);
    *(float4*)(S + (size_t)row * ldS + c4) = v;   // 528B/576B row pitch: 16B ok
  }
}

// Block-level 128x128 @ 128x128 -> global D.  256 threads = 8 waves.
// Wave w owns fragment grid rows {2*(w>>1), 2*(w>>1)+1}, cols 4*(w&1)+[0,4).
template <bool NT>
__device__ __forceinline__ void block_matmul(
    const float* __restrict__ L, const float* __restrict__ Rg,
    float* __restrict__ D, float* Ls, float* Rs)
{
  const int tid = threadIdx.x;
  stage_matrix(L,  Ls, LDA, tid);
  stage_matrix(Rg, Rs, LDB, tid);
  __syncthreads();
  const int wave = tid >> 5, lane = tid & 31;
  const int tmBase = (wave >> 1) * 2;
  const int tnBase = (wave & 1) * 4;
  v8f zero = {};
  v8f acc[8];
#pragma unroll
  for (int i = 0; i < 8; ++i) acc[i] = zero;
  wave_mm_2x4(Ls, LDA, Rs, LDB, tmBase, tnBase, lane, acc);
  wave_store_2x4<NT>(D, MAT, tmBase, tnBase, lane, acc);
}

// ---- kernel 1: expm(P - P^T) via scaling (2^-10) + 12-term Taylor + 10 sq --
__global__ void __launch_bounds__(256) expm_kernel(
    const float* __restrict__ Praw, float* __restrict__ prim)
{
  extern __shared__ float sm[];
  float* A = sm;                  // scaled skew input
  float* X = sm + LDSMAT_E;       // current Taylor term
  float* S = sm + 2 * LDSMAT_E;   // accumulated sum / result
  const float* P = Praw + (size_t)blockIdx.x * MATSZ;
  const int tid = threadIdx.x;
  const float sc = 1.0f / 1024.0f;    // 2^-10
  for (int idx = tid; idx < MATSZ; idx += 256) {
    const int i = idx >> 7, j = idx & 127;
    const float a = (P[i * MAT + j] - P[j * MAT + i]) * sc;
    A[i * LDSE + j] = a;
    X[i * LDSE + j] = a;
    S[i * LDSE + j] = a + ((i == j) ? 1.0f : 0.0f);
  }
  __syncthreads();
  const int wave = tid >> 5, lane = tid & 31;
  const int tmBase = (wave >> 1) * 2;
  const int tnBase = (wave & 1) * 4;
  v8f zero = {};
  // Taylor terms 2..12:  X <- (X @ A) / t ;  S += X
  for (int term = 2; term <= 12; ++term) {
    v8f acc[8];
#pragma unroll
    for (int i = 0; i < 8; ++i) acc[i] = zero;
    wave_mm_2x4(X, LDSE, A, LDSE, tmBase, tnBase, lane, acc);
    __syncthreads();                     // all reads of X complete
    const float inv = 1.0f / (float)term;
#pragma unroll
    for (int i = 0; i < 8; ++i) acc[i] = acc[i] * inv;
    wave_store_2x4<false>(X, LDSE, tmBase, tnBase, lane, acc);
    __syncthreads();
    for (int idx = tid; idx < MATSZ; idx += 256) {
      const int i = idx >> 7, j = idx & 127;
      S[i * LDSE + j] += X[i * LDSE + j];
    }
    __syncthreads();
  }
  // 10 squarings:  S <- S @ S
  for (int r = 0; r < 10; ++r) {
    v8f acc[8];
#pragma unroll
    for (int i = 0; i < 8; ++i) acc[i] = zero;
    wave_mm_2x4(S, LDSE, S, LDSE, tmBase, tnBase, lane, acc);
    __syncthreads();                     // all reads of S complete
    wave_store_2x4<false>(S, LDSE, tmBase, tnBase, lane, acc);
    __syncthreads();
  }
  float* out = prim + (size_t)blockIdx.x * MATSZ;
  for (int idx = tid; idx < MATSZ; idx += 256) {
    const int i = idx >> 7, j = idx & 127;
    out[idx] = S[i * LDSE + j];
  }
}

// ---- kernel 2: Tab[0] = I ; Tab[1+b] = W_b^T -------------------------------
__global__ void __launch_bounds__(256) init_kernel(
    const float* __restrict__ prim, float* __restrict__ U)
{
  const int b = blockIdx.x;
  const int tid = threadIdx.x;
  if (b == 0) {
    for (int idx = tid; idx < MATSZ; idx += 256)
      U[idx] = ((idx >> 7) == (idx & 127)) ? 1.0f : 0.0f;
  } else {
    const float* P = prim + (size_t)(b - 1) * MATSZ;
    float* Dm = U + (size_t)b * MATSZ;
    for (int idx = tid; idx < MATSZ; idx += 256) {
      const int i = idx >> 7, j = idx & 127;
      Dm[idx] = P[j * MAT + i];          // transpose
    }
  }
}

// ---- kernel 3: tree DP level k: Tab[q-1] = Wt[q&1] @ Tab[(q>>1)-1] ---------
__global__ void __launch_bounds__(256) level_kernel(
    float* __restrict__ U, int level)
{
  extern __shared__ float sm[];
  float* Ls = sm;
  float* Rs = sm + LDSMAT_A;
  const int q = (1 << level) + blockIdx.x;
  const float* L = U + (size_t)(1 + (q & 1)) * MATSZ;  // Wt[q&1]
  const float* R = U + (size_t)((q >> 1) - 1) * MATSZ;
  float*       D = U + (size_t)(q - 1) * MATSZ;
  block_matmul<false>(L, R, D, Ls, Rs);
}

// ---- kernel 4: per-position single matmul, NT stores of 537MB output -------
__global__ void __launch_bounds__(256) final_kernel(
    const int* __restrict__ unique, const float* __restrict__ U,
    float* __restrict__ out)
{
  extern __shared__ float sm[];
  float* Ls = sm;
  float* Rs = sm + LDSMAT_A;
  const int pos = unique[blockIdx.x];
  int li, ri;
  if (pos >= 256) { li = 255 + (pos & 255); ri = (pos >> 8) - 1; }
  else            { li = pos - 1;           ri = 0; }         // Tab[0] = I
  const float* L = U + (size_t)li * MATSZ;
  const float* R = U + (size_t)ri * MATSZ;
  float*       D = out + (size_t)blockIdx.x * MATSZ;
  block_matmul<true>(L, R, D, Ls, Rs);
}

// ---------------------------------------------------------------------------
extern "C" void kernel_launch(void* const* d_in, const int* in_sizes, int n_in,
                              void* d_out, int out_size, void* d_ws, size_t ws_size,
                              hipStream_t stream)
{
  (void)in_sizes; (void)n_in; (void)out_size; (void)ws_size;
  const int*   unique = (const int*)d_in[0];
  const float* praw   = (const float*)d_in[1];   // [2,128,128] f32
  // d_in[2] (identity) is not needed.

  float* U    = (float*)d_ws;                    // 511 matrices (33.5 MB)
  float* prim = U + (size_t)511 * MATSZ;         // 2 matrices

  const size_t sm_mm   = (size_t)(LDSMAT_A + LDSMAT_B) * sizeof(float); // 141312
  const size_t sm_expm = (size_t)3 * LDSMAT_E * sizeof(float);          // 221184

  hipLaunchKernelGGL(expm_kernel, dim3(2), dim3(256), sm_expm, stream,
                     praw, prim);
  hipLaunchKernelGGL(init_kernel, dim3(3), dim3(256), 0, stream, prim, U);
  for (int k = 2; k <= 8; ++k)
    hipLaunchKernelGGL(level_kernel, dim3(1u << k), dim3(256), sm_mm, stream,
                       U, k);
  hipLaunchKernelGGL(final_kernel, dim3(8192), dim3(256), sm_mm, stream,
                     unique, U, (float*)d_out);
}